// Aggregator_45286135169721
// MI455X (gfx1250) — compile-verified
//
#include <hip/hip_runtime.h>
#include <hip/hip_bf16.h>
#include <stdint.h>

typedef float v4f __attribute__((ext_vector_type(4)));

#define D        128   // feature dim (floats per row)
#define S        25    // samples per node
#define CHUNK    5     // rows per async chunk
#define NCHUNK   5     // S / CHUNK
#define WAVES_PB 4
#define BLOCK    (WAVES_PB * 32)

__global__ __launch_bounds__(BLOCK) void
gather_mean_async(const float* __restrict__ feat,
                  const int*   __restrict__ nbr,
                  float* __restrict__ out,
                  int n_nodes)
{
    // per-wave double-buffered row staging: 4 waves * 2 stages * 5 rows * 512B = 20480B
    __shared__ v4f smem[WAVES_PB][2][CHUNK][32];

    const int lane = threadIdx.x & 31;
    const int wave = threadIdx.x >> 5;
    const unsigned long long fbase = (unsigned long long)(uintptr_t)feat;
    const int waves_total = gridDim.x * WAVES_PB;
    const unsigned lanebytes = (unsigned)lane * 16u;

    // wave-uniform node id -> index loads become scalar (SMEM) loads
    int node = __builtin_amdgcn_readfirstlane((int)(blockIdx.x * WAVES_PB + wave));

    v4f* __restrict__ outv = (v4f*)out;

    for (; node < n_nodes; node += waves_total) {
        // 25 indices at a uniform address: lowers to s_load clauses (KMcnt),
        // values live in SGPRs -> no ds_bpermute broadcast needed.
        int idxs[S];
#pragma unroll
        for (int s = 0; s < S; ++s)
            idxs[s] = nbr[(long long)node * S + s];

        v4f acc = {0.f, 0.f, 0.f, 0.f};

        // Issue one chunk of 5 async row-gathers (global -> LDS), 512B/row,
        // lane l moves bytes [16l,16l+16) of the row. GVS addressing:
        // mem = saddr(features) + vgpr(idx*512 + lane*16).
#define ISSUE_CHUNK(cc, stage)                                                  \
        {                                                                       \
            _Pragma("unroll")                                                   \
            for (int r = 0; r < CHUNK; ++r) {                                   \
                unsigned voff = ((unsigned)idxs[(cc) * CHUNK + r] << 9) + lanebytes; \
                unsigned ldsa = (unsigned)(uintptr_t)&smem[wave][(stage)][r][lane]; \
                asm volatile("global_load_async_to_lds_b128 %0, %1, %2"         \
                             :: "v"(ldsa), "v"(voff), "s"(fbase) : "memory");   \
            }                                                                   \
        }

        // prologue: make sure prior node's ds reads of stage0 are done, then fill it
        asm volatile("s_wait_dscnt 0x0" ::: "memory");
        ISSUE_CHUNK(0, 0)

#pragma unroll
        for (int c = 0; c < NCHUNK; ++c) {
            if (c + 1 < NCHUNK) {
                // stage (c+1)&1 was last read by chunk c-1 (for c>0); drain ds reads
                // before the async engine overwrites it, then prefetch next chunk.
                asm volatile("s_wait_dscnt 0x0" ::: "memory");
                ISSUE_CHUNK(c + 1, (c + 1) & 1)
                // <=5 outstanding => chunk c (oldest 5, in-order retire) landed in LDS
                asm volatile("s_wait_asynccnt 0x5" ::: "memory");
            } else {
                asm volatile("s_wait_asynccnt 0x0" ::: "memory");
            }
#pragma unroll
            for (int r = 0; r < CHUNK; ++r) {
                acc += smem[wave][c & 1][r][lane];   // ds_load_b128 + adds
            }
        }
#undef ISSUE_CHUNK

        acc *= (1.0f / (float)S);
        // Streaming store: keep the 102MB feature table resident in the 192MB L2,
        // don't let the 51MB output evict it.
        __builtin_nontemporal_store(acc, &outv[(long long)node * (D / 4) + lane]);
    }
}

extern "C" void kernel_launch(void* const* d_in, const int* in_sizes, int n_in,
                              void* d_out, int out_size, void* d_ws, size_t ws_size,
                              hipStream_t stream)
{
    (void)n_in; (void)d_ws; (void)ws_size; (void)in_sizes;
    const float* feat = (const float*)d_in[0];
    const int*   nbr  = (const int*)d_in[1];
    float*       out  = (float*)d_out;

    int n_nodes = out_size / D;            // 100000 for the reference shapes
    int blocks  = (n_nodes + WAVES_PB - 1) / WAVES_PB;
    if (blocks > 8192) blocks = 8192;      // grid-stride covers the rest
    if (blocks < 1)    blocks = 1;

    hipLaunchKernelGGL(gather_mean_async, dim3(blocks), dim3(BLOCK), 0, stream,
                       feat, nbr, out, n_nodes);
}